// EmformerEncoderLayer_17111149707922
// MI455X (gfx1250) — compile-verified
//
#include <hip/hip_runtime.h>
#include <hip/hip_bf16.h>

// Emformer conv module for MI455X (gfx1250, wave32, WMMA, async-to-LDS).
// Shapes: U=2048, R=512, B=16, D=512, T=2560, K=31 taps, CACHE=30.
#define U_LEN 2048
#define R_LEN 512
#define B_SZ  16
#define D_SZ  512
#define T_LEN (U_LEN + R_LEN)    // 2560
#define KSZ   31
#define CS    30
#define RCL   8
#define CL    32

#define COL_STRIDE 516           // padded LDS column stride (516 % 64 == 4)

typedef __attribute__((ext_vector_type(2))) float v2f;
typedef __attribute__((ext_vector_type(8))) float v8f;
typedef __attribute__((ext_vector_type(4))) int   v4i;

typedef __attribute__((address_space(1))) v4i* gas_v4i_ptr;  // global
typedef __attribute__((address_space(3))) v4i* las_v4i_ptr;  // LDS

__device__ __forceinline__ float fast_sigmoid(float x) {
    return 1.0f / (1.0f + __expf(-x));
}

// 128-bit global -> LDS copy; async (ASYNCcnt-tracked) when the gfx1250
// builtin exists, synchronous fallback otherwise.
__device__ __forceinline__ void copy_b128_to_lds(const float* g, float* l) {
#if __has_builtin(__builtin_amdgcn_global_load_async_to_lds_b128)
    __builtin_amdgcn_global_load_async_to_lds_b128(
        (gas_v4i_ptr)(v4i*)const_cast<float*>(g),
        (las_v4i_ptr)(v4i*)l, 0, 0);
#else
    *(float4*)l = *(const float4*)g;
#endif
}

__device__ __forceinline__ void wait_async_copies() {
#if __has_builtin(__builtin_amdgcn_s_wait_asynccnt)
    __builtin_amdgcn_s_wait_asynccnt(0);
#else
    asm volatile("s_wait_asynccnt 0" ::: "memory");
#endif
}

// ---------------------------------------------------------------------------
// Kernel 1: pointwise conv w1 (GEMM 1024x512 @ 512x(B*T)) + bias + GLU.
// grid = (T/2, 4), block = 256 (8 wave32). A block stages the activation
// tile for 2 time steps (32 columns x K=512, 64.5KB LDS) once; each wave
// computes one channel-pair tile (a-half + g-half) for BOTH time tiles:
// per K=4 step -> 2 global b64 (A) + 2 ds b64 (B) + 4 WMMAs.
// Output: glu[t][b][d]  (T,B,D) fp32 in workspace.
// ---------------------------------------------------------------------------
__global__ void __launch_bounds__(256) pw1_glu_kernel(
    const float* __restrict__ utt, const float* __restrict__ rc,
    const float* __restrict__ w1,  const float* __restrict__ b1,
    float* __restrict__ glu)
{
    __shared__ float xs[32 * COL_STRIDE];      // 2 time slabs, padded columns

    const int t0   = blockIdx.x * 2;           // t0, t0+1 (never straddles R)
    const int wave = threadIdx.x >> 5;         // 0..7
    const int lane = threadIdx.x & 31;
    const int cBase = (blockIdx.y * 8 + wave) * 16;  // 32 c-pairs cover D=512

    // ---- cooperative stage: two contiguous 32KB slabs -> LDS (b128) -------
    const float* slab0 = (t0 < R_LEN)
        ? (rc  + (size_t)t0 * B_SZ * D_SZ)
        : (utt + (size_t)(t0 - R_LEN) * B_SZ * D_SZ);
    const float* slab1 = ((t0 + 1) < R_LEN)
        ? (rc  + (size_t)(t0 + 1) * B_SZ * D_SZ)
        : (utt + (size_t)(t0 + 1 - R_LEN) * B_SZ * D_SZ);

    for (int i = threadIdx.x; i < 4096; i += 256) {   // 4096 float4 pieces
        const int col = i >> 7;                        // 0..31
        const int off = (i & 127) * 4;                 // 0..508
        const float* src = ((col < 16) ? slab0 + (size_t)col * D_SZ
                                       : slab1 + (size_t)(col - 16) * D_SZ) + off;
        copy_b128_to_lds(src, &xs[col * COL_STRIDE + off]);
    }
    wait_async_copies();
    __syncthreads();

    // ---- WMMA mainloop -----------------------------------------------------
    const int m    = lane & 15;                // A-row / B-col index
    const int hi   = lane >> 4;
    const int koff = hi << 1;                  // K sub-offset {0,2}

    const float* wa = w1 + (size_t)(cBase + m)        * D_SZ;  // a-half row
    const float* wg = w1 + (size_t)(cBase + m + D_SZ) * D_SZ;  // g-half row
    const float* bs0 = &xs[(0 + m) * COL_STRIDE + koff];       // t0 column
    const float* bs1 = &xs[(16 + m) * COL_STRIDE + koff];      // t0+1 column

    v8f acc_a0 = {}, acc_a1 = {}, acc_g0 = {}, acc_g1 = {};
    #pragma unroll 4
    for (int k = 0; k < D_SZ; k += 4) {
        v2f a   = *(const v2f*)(wa + k + koff);
        v2f g   = *(const v2f*)(wg + k + koff);
        v2f bt0 = *(const v2f*)(bs0 + k);      // ds_load_b64
        v2f bt1 = *(const v2f*)(bs1 + k);      // ds_load_b64
        acc_a0 = __builtin_amdgcn_wmma_f32_16x16x4_f32(
            false, a, false, bt0, (short)0, acc_a0, false, false);
        acc_a1 = __builtin_amdgcn_wmma_f32_16x16x4_f32(
            false, a, false, bt1, (short)0, acc_a1, false, false);
        acc_g0 = __builtin_amdgcn_wmma_f32_16x16x4_f32(
            false, g, false, bt0, (short)0, acc_g0, false, false);
        acc_g1 = __builtin_amdgcn_wmma_f32_16x16x4_f32(
            false, g, false, bt1, (short)0, acc_g1, false, false);
    }

    // ---- epilogue: bias + GLU, C/D element (M = v + 8*hi, N = m) ----------
    #pragma unroll
    for (int tt = 0; tt < 2; ++tt) {
        const v8f& aa = tt ? acc_a1 : acc_a0;
        const v8f& gg = tt ? acc_g1 : acc_g0;
        float* outp = glu + ((size_t)(t0 + tt) * B_SZ + m) * D_SZ + cBase + hi * 8;
        #pragma unroll
        for (int v = 0; v < 8; ++v) {
            const int ch = cBase + hi * 8 + v;
            const float av = aa[v] + b1[ch];
            const float gv = gg[v] + b1[ch + D_SZ];
            outp[v] = av * fast_sigmoid(gv);   // GLU fused into epilogue
        }
    }
}

// ---------------------------------------------------------------------------
// Kernel 2: depthwise 31-tap conv (groups=D) over both the utterance path
// (with cache prefix) and the 64 right-context chunks (gathered padding),
// fused with bias + x*sigmoid(x-1).  Output conv[t][b][d] (T,B,D).
// ---------------------------------------------------------------------------
__global__ void __launch_bounds__(256) dwconv_silu_kernel(
    const float* __restrict__ glu, const float* __restrict__ cache,
    const float* __restrict__ wd,  const float* __restrict__ bd,
    float* __restrict__ conv)
{
    const int idx = blockIdx.x * 256 + threadIdx.x;     // over T*B*D
    const int d = idx % D_SZ;
    const int b = (idx / D_SZ) % B_SZ;
    const int t = idx / (D_SZ * B_SZ);

    const size_t colStride = (size_t)B_SZ * D_SZ;
    const float* gcol = glu + (size_t)b * D_SZ + d;           // + time*colStride
    const float* ccol = cache + ((size_t)b * D_SZ + d) * CS;  // (B,D,30)
    const float* wrow = wd + (size_t)d * KSZ;

    float acc = 0.f;
    if (t >= R_LEN) {
        // utterance path: pad_utt = cache(30) ++ glu[R..R+U)
        const int ot = t - R_LEN;
        #pragma unroll
        for (int k = 0; k < KSZ; ++k) {
            const int j = ot + k;                              // pad_utt index
            const float xv = (j < CS) ? ccol[j]
                                      : gcol[(size_t)(R_LEN + j - CS) * colStride];
            acc = fmaf(xv, wrow[k], acc);
        }
    } else {
        // right-context path: chunk n, 30 gathered pad frames ++ 8 rc frames
        const int n = t >> 3;                  // chunk 0..63
        const int r = t & 7;
        const int start = (n + 1) * CL;        // = min(2048,(n+1)*32) exactly
        #pragma unroll
        for (int k = 0; k < KSZ; ++k) {
            const int i = r + k;               // 0..37 within [pad(30) | rc(8)]
            float xv;
            if (i < CS) {
                const int j = start + i;       // pad_utt index (>=32 -> glu)
                xv = (j < CS) ? ccol[j]
                              : gcol[(size_t)(R_LEN + j - CS) * colStride];
            } else {
                xv = gcol[(size_t)(n * RCL + (i - CS)) * colStride];  // rc frames
            }
            acc = fmaf(xv, wrow[k], acc);
        }
    }
    const float v = acc + bd[d];
    conv[idx] = v * fast_sigmoid(v - 1.0f);    // x*sigmoid(x-1) fused
}

// ---------------------------------------------------------------------------
// Kernel 3: new_cache = last 30 frames of pad_utt (pre-depthwise GLU output).
// new_cache[b][d][i] = glu[time R+U-30+i][b][d]
// ---------------------------------------------------------------------------
__global__ void __launch_bounds__(256) new_cache_kernel(
    const float* __restrict__ glu, float* __restrict__ out_cache)
{
    const int idx = blockIdx.x * 256 + threadIdx.x;
    if (idx >= B_SZ * D_SZ * CS) return;
    const int i = idx % CS;
    const int d = (idx / CS) % D_SZ;
    const int b = idx / (CS * D_SZ);
    out_cache[idx] =
        glu[((size_t)(R_LEN + U_LEN - CS + i) * B_SZ + b) * D_SZ + d];
}

// ---------------------------------------------------------------------------
// Kernel 4: pointwise conv w2 (GEMM 512x512 @ 512x(B*T)) + bias, scattered
// straight into the (U,B,D) / (R,B,D) output regions. Same LDS-staged
// structure as kernel 1: per K=4 step -> 1 global b64 + 2 ds b64 + 2 WMMAs.
// ---------------------------------------------------------------------------
__global__ void __launch_bounds__(256) pw2_kernel(
    const float* __restrict__ conv, const float* __restrict__ w2,
    const float* __restrict__ b2,
    float* __restrict__ out_utt, float* __restrict__ out_rc)
{
    __shared__ float xs[32 * COL_STRIDE];

    const int t0   = blockIdx.x * 2;
    const int wave = threadIdx.x >> 5;
    const int lane = threadIdx.x & 31;
    const int cBase = (blockIdx.y * 8 + wave) * 16;   // 32 tiles cover D=512

    const float* slab0 = conv + (size_t)t0 * B_SZ * D_SZ;
    for (int i = threadIdx.x; i < 4096; i += 256) {
        const int col = i >> 7;
        const int off = (i & 127) * 4;
        copy_b128_to_lds(slab0 + (size_t)col * D_SZ + off,
                         &xs[col * COL_STRIDE + off]);
    }
    wait_async_copies();
    __syncthreads();

    const int m    = lane & 15;
    const int hi   = lane >> 4;
    const int koff = hi << 1;

    const float* wrow = w2 + (size_t)(cBase + m) * D_SZ;
    const float* bs0 = &xs[(0 + m) * COL_STRIDE + koff];
    const float* bs1 = &xs[(16 + m) * COL_STRIDE + koff];

    v8f acc0 = {}, acc1 = {};
    #pragma unroll 4
    for (int k = 0; k < D_SZ; k += 4) {
        v2f a   = *(const v2f*)(wrow + k + koff);
        v2f bt0 = *(const v2f*)(bs0 + k);
        v2f bt1 = *(const v2f*)(bs1 + k);
        acc0 = __builtin_amdgcn_wmma_f32_16x16x4_f32(
            false, a, false, bt0, (short)0, acc0, false, false);
        acc1 = __builtin_amdgcn_wmma_f32_16x16x4_f32(
            false, a, false, bt1, (short)0, acc1, false, false);
    }

    #pragma unroll
    for (int tt = 0; tt < 2; ++tt) {
        const int t = t0 + tt;
        const v8f& acc = tt ? acc1 : acc0;
        float* dst = (t < R_LEN)
            ? (out_rc  + ((size_t)t * B_SZ + m) * D_SZ)
            : (out_utt + ((size_t)(t - R_LEN) * B_SZ + m) * D_SZ);
        dst += cBase + hi * 8;
        #pragma unroll
        for (int v = 0; v < 8; ++v)
            dst[v] = acc[v] + b2[cBase + hi * 8 + v];
    }
}

// ---------------------------------------------------------------------------
extern "C" void kernel_launch(void* const* d_in, const int* in_sizes, int n_in,
                              void* d_out, int out_size, void* d_ws, size_t ws_size,
                              hipStream_t stream)
{
    const float* utt   = (const float*)d_in[0];  // (U,B,D)
    const float* rc    = (const float*)d_in[1];  // (R,B,D)
    const float* cache = (const float*)d_in[2];  // (B,D,30)
    const float* w1    = (const float*)d_in[3];  // (2D,D,1)
    const float* b1    = (const float*)d_in[4];  // (2D)
    const float* wd    = (const float*)d_in[5];  // (D,1,31)
    const float* bd    = (const float*)d_in[6];  // (D)
    const float* w2    = (const float*)d_in[7];  // (D,D,1)
    const float* b2    = (const float*)d_in[8];  // (D)

    float* out       = (float*)d_out;
    float* out_utt   = out;                                     // (U,B,D)
    float* out_rc    = out + (size_t)U_LEN * B_SZ * D_SZ;       // (R,B,D)
    float* out_cache = out_rc + (size_t)R_LEN * B_SZ * D_SZ;    // (B,D,30)

    float* glu  = (float*)d_ws;                                 // (T,B,D) 80MB
    float* conv = glu + (size_t)T_LEN * B_SZ * D_SZ;            // (T,B,D) 80MB

    const dim3 blk(256);
    pw1_glu_kernel<<<dim3(T_LEN / 2, 4), blk, 0, stream>>>(utt, rc, w1, b1, glu);
    dwconv_silu_kernel<<<dim3((T_LEN * B_SZ * D_SZ) / 256), blk, 0, stream>>>(
        glu, cache, wd, bd, conv);
    new_cache_kernel<<<dim3((B_SZ * D_SZ * CS + 255) / 256), blk, 0, stream>>>(
        glu, out_cache);
    pw2_kernel<<<dim3(T_LEN / 2, 4), blk, 0, stream>>>(conv, w2, b2, out_utt, out_rc);
}